// BilinearDecoder_79001628443243
// MI455X (gfx1250) — compile-verified
//
#include <hip/hip_runtime.h>
#include <math.h>

// CDNA5 (gfx1250) wave32 WMMA types
typedef __attribute__((ext_vector_type(2))) float v2f;
typedef __attribute__((ext_vector_type(8))) float v8f;

#define WDIM   512       // feature dimension (K and N of the GEMM)
#define APITCH 516       // padded LDS row pitch (floats): %64==4 -> conflict-free,
                         // *4 bytes = 2064 B, 16B-aligned for B128 async chunks

// Truncate a generic pointer to the wave-relative LDS byte address
// (flat LDS aperture: addr[31:0] is the group-segment offset).
__device__ __forceinline__ unsigned lds_addr32(const void* p) {
    return (unsigned)(uintptr_t)p;
}

// ---------------------------------------------------------------------------
// Phase 0: Wt = W^T (512x512).  Makes B fragments contiguous 8B loads in the
// WMMA K-loop (one global_load_b64 instead of two strided b32 per fragment).
// ---------------------------------------------------------------------------
__global__ __launch_bounds__(256) void transpose_w(
    const float* __restrict__ W, float* __restrict__ Wt)
{
    __shared__ float tile[32][33];            // +1 pad: conflict-free transpose
    const int bx = blockIdx.x & 15;           // 16 col-tiles
    const int by = blockIdx.x >> 4;           // 16 row-tiles
    const int tx = threadIdx.x & 31;
    const int ty = threadIdx.x >> 5;          // 0..7

    #pragma unroll
    for (int j = 0; j < 32; j += 8)
        tile[ty + j][tx] = W[(size_t)(by * 32 + ty + j) * WDIM + bx * 32 + tx];
    __syncthreads();
    #pragma unroll
    for (int j = 0; j < 32; j += 8)
        Wt[(size_t)(bx * 32 + ty + j) * WDIM + by * 32 + tx] = tile[tx][ty + j];
}

// ---------------------------------------------------------------------------
// Phase 1: Y = Z @ W   (M x 512) = (M x 512)(512 x 512), fp32 WMMA 16x16x4.
// One block (256 thr = 8 waves) per 16-row M tile; wave w owns N = [w*64, w*64+64).
// The shared 16x512 A-stripe is staged into LDS once via async-to-LDS loads,
// removing the 8x-redundant per-wave A traffic from L2.  B comes from Wt so
// each fragment is a single aligned global_load_b64.
// ---------------------------------------------------------------------------
__global__ __launch_bounds__(256) void gemm_zw_f32_wmma(
    const float* __restrict__ Z,
    const float* __restrict__ Wt,     // transposed weight [n][k]
    float* __restrict__ Y)
{
    __shared__ __align__(16) float Atile[16 * APITCH];   // ~33 KB of 320 KB LDS

    const int t    = threadIdx.x;
    const int lane = t & 31;
    const int wave = t >> 5;                  // 0..7
    const int mt   = blockIdx.x;              // 16-row tile index
    const int hi   = lane >> 4;               // 0 or 1 (lane half)
    const int mr   = lane & 15;               // A-row within tile / B,N col
    const int nbase = wave * 64;

    // ---- async stage A stripe: 16 rows x 512 floats = 2048 x 16B chunks ----
    {
        const float* gbase = Z + (size_t)mt * 16 * WDIM;
        #pragma unroll
        for (int j = 0; j < 8; ++j) {
            const int id = j * 256 + t;           // chunk id 0..2047
            const int r  = id >> 7;               // row 0..15 (128 chunks/row)
            const int c  = (id & 127) << 2;       // float col, step 4
            const unsigned laddr = lds_addr32(&Atile[r * APITCH + c]);
            const float*   gaddr = gbase + (size_t)r * WDIM + c;
            asm volatile("global_load_async_to_lds_b128 %0, %1, off"
                         :: "v"(laddr), "v"(gaddr) : "memory");
        }
        asm volatile("s_wait_asynccnt 0" ::: "memory");
        __syncthreads();
    }

    // per-lane B base: row n = nbase+mr of Wt, contiguous in K
    const float* Wn = Wt + (size_t)(nbase + mr) * WDIM;

    v8f c0 = {}; v8f c1 = {}; v8f c2 = {}; v8f c3 = {};

    #pragma unroll 4
    for (int k = 0; k < WDIM; k += 4) {
        const int ka = k + 2 * hi;            // this lane-half's K pair base
        // A fragment from LDS: A[mr][ka], A[mr][ka+1] (ds_load_2addr_b64 pairs)
        v2f a = *(const v2f*)(&Atile[mr * APITCH + ka]);

        // B fragments: single 8B loads from transposed W (Wt[n][ka], Wt[n][ka+1])
        v2f b0 = *(const v2f*)(Wn + 0 * 16 * WDIM + ka);
        v2f b1 = *(const v2f*)(Wn + 1 * 16 * WDIM + ka);
        v2f b2 = *(const v2f*)(Wn + 2 * 16 * WDIM + ka);
        v2f b3 = *(const v2f*)(Wn + 3 * 16 * WDIM + ka);

        // 8 args: (neg_a, A, neg_b, B, c_mod, C, reuse_a, reuse_b)
        c0 = __builtin_amdgcn_wmma_f32_16x16x4_f32(false, a, false, b0, (short)0, c0, false, false);
        c1 = __builtin_amdgcn_wmma_f32_16x16x4_f32(false, a, false, b1, (short)0, c1, false, false);
        c2 = __builtin_amdgcn_wmma_f32_16x16x4_f32(false, a, false, b2, (short)0, c2, false, false);
        c3 = __builtin_amdgcn_wmma_f32_16x16x4_f32(false, a, false, b3, (short)0, c3, false, false);
    }

    // C/D layout: VGPR j, lane L -> M = j + 8*(L/16), N = L%16
    float* Yt = Y + (size_t)(mt * 16 + 8 * hi) * WDIM + nbase + mr;
    #pragma unroll
    for (int j = 0; j < 8; ++j) {
        Yt[(size_t)j * WDIM + 0]  = c0[j];
        Yt[(size_t)j * WDIM + 16] = c1[j];
        Yt[(size_t)j * WDIM + 32] = c2[j];
        Yt[(size_t)j * WDIM + 48] = c3[j];
    }
}

// ---------------------------------------------------------------------------
// Phase 2: score_e = sigmoid( Y[row_e] . Z[col_e] ).  One wave32 per edge.
// Rows live in L2 (Z + Y = 41 MB << 192 MB L2), so gathers are L2 hits.
// ---------------------------------------------------------------------------
__global__ __launch_bounds__(256) void edge_bilinear_score(
    const float* __restrict__ Y,
    const float* __restrict__ Z,
    const int* __restrict__ rows,
    const int* __restrict__ cols,
    float* __restrict__ out,
    int E)
{
    const int e    = (int)((blockIdx.x * blockDim.x + threadIdx.x) >> 5);
    const int lane = threadIdx.x & 31;
    if (e >= E) return;

    const float* yr = Y + (size_t)rows[e] * WDIM;
    const float* zc = Z + (size_t)cols[e] * WDIM;

    float s = 0.0f;
    #pragma unroll
    for (int q = 0; q < 4; ++q) {
        const int i = lane * 4 + q * 128;     // 128B-coalesced float4 per lane
        float4 a = *(const float4*)(yr + i);
        float4 b = *(const float4*)(zc + i);
        s += a.x * b.x + a.y * b.y + a.z * b.z + a.w * b.w;
    }

    // wave32 tree reduction
    #pragma unroll
    for (int off = 16; off > 0; off >>= 1)
        s += __shfl_xor(s, off, 32);

    if (lane == 0)
        out[e] = 1.0f / (1.0f + __expf(-s));
}

// ---------------------------------------------------------------------------
extern "C" void kernel_launch(void* const* d_in, const int* in_sizes, int n_in,
                              void* d_out, int out_size, void* d_ws, size_t ws_size,
                              hipStream_t stream)
{
    const float* Z     = (const float*)d_in[0];   // [10000, 512] fp32
    const float* W     = (const float*)d_in[1];   // [512, 512]   fp32
    const int*   edges = (const int*)  d_in[2];   // [2, E] int32
    float*       out   = (float*)d_out;           // [E] fp32

    const int M = in_sizes[0] / WDIM;             // 10000 (multiple of 16)
    const int E = out_size;                       // 200000

    // workspace layout: Y [M*512] floats, then Wt [512*512] floats (~21.5 MB)
    float* Ybuf = (float*)d_ws;
    float* Wt   = Ybuf + (size_t)M * WDIM;

    // Phase 0: transpose W (256 blocks of 256 threads, 32x32 LDS tiles)
    transpose_w<<<256, 256, 0, stream>>>(W, Wt);

    // Phase 1: Y = Z @ W  via fp32 WMMA
    gemm_zw_f32_wmma<<<M / 16, 256, 0, stream>>>(Z, Wt, Ybuf);

    // Phase 2: per-edge bilinear score + sigmoid (wave per edge, 8 edges/block)
    const int blocks = (E + 7) / 8;
    edge_bilinear_score<<<blocks, 256, 0, stream>>>(Ybuf, Z, edges, edges + E, out, E);
}